// GATBlock_1417339207724
// MI455X (gfx1250) — compile-verified
//
#include <hip/hip_runtime.h>
#include <hip/hip_bf16.h>
#include <math.h>

// ---- CDNA5 (gfx1250) GAT block: WMMA f32 GEMM + edge softmax scatter + BN ----
// wave32 everywhere. Matrix op uses V_WMMA_F32_16X16X4_F32 (exact f32, K-loop).

typedef __attribute__((ext_vector_type(2))) float v2f;
typedef __attribute__((ext_vector_type(8))) float v8f;

#define FEAT 128
#define HNUM 4
#define CNUM 32
#define NEG_SLOPE 0.2f
#define BN_EPS 1e-5f

// Monotonic float -> signed-int key (for integer atomicMax as float max)
__device__ __forceinline__ int ordkey(float f) {
    int i = __float_as_int(f);
    return (i >= 0) ? i : ~(i & 0x7FFFFFFF);
}
__device__ __forceinline__ float ordval(int k) {
    return (k >= 0) ? __int_as_float(k) : __int_as_float((~k) | 0x80000000);
}
__device__ __forceinline__ float lrelu(float z) { return z > 0.f ? z : NEG_SLOPE * z; }

// ---------------------------------------------------------------------------
// Kernel 1: h = x @ W  via V_WMMA_F32_16X16X4_F32.
// One block = 16-row tile of x staged in LDS; 8 waves each own a 16-col slice.
// A (16x4 f32): lanes 0-15 hold K=0 (v0) / K=1 (v1); lanes 16-31 hold K=2/K=3.
// B (4x16 f32): lanes 0-15 hold K=0/K=1; lanes 16-31 hold K=2/K=3 (N = lane&15).
// C/D (16x16 f32, 8 VGPRs): lane<16 -> M=v, N=lane; lane>=16 -> M=v+8, N=lane-16.
// ---------------------------------------------------------------------------
#define XS_STRIDE 132   // 128 + 4 dword pad: LDS reads land on distinct banks
__global__ __launch_bounds__(256) void gemm_wmma_f32(const float* __restrict__ x,
                                                     const float* __restrict__ W,
                                                     float* __restrict__ h, int n) {
    __shared__ float xs[16 * XS_STRIDE];

    const int row0 = blockIdx.x * 16;
    const int tid  = threadIdx.x;

    // Cooperative coalesced load of the 16x128 x-tile (float4 per thread x2).
    for (int i = tid; i < 16 * FEAT / 4; i += 256) {
        int r = (i * 4) / FEAT;
        int c = (i * 4) % FEAT;
        int rr = row0 + r; if (rr >= n) rr = n - 1;          // clamp (tail-safe)
        float4 v = *(const float4*)(x + (size_t)rr * FEAT + c);
        *(float4*)(xs + r * XS_STRIDE + c) = v;
    }
    __syncthreads();

    const int wave = tid >> 5;           // 8 waves (wave32)
    const int lane = tid & 31;
    const int lmod = lane & 15;
    const int hi   = (lane & 16) ? 1 : 0;
    const int col0 = wave * 16;

    v8f acc = {};
    #pragma unroll
    for (int k = 0; k < FEAT; k += 4) {
        const int ka = k + (hi ? 2 : 0);
        v2f a, b;
        a.x = xs[lmod * XS_STRIDE + ka];
        a.y = xs[lmod * XS_STRIDE + ka + 1];
        b.x = W[(size_t)ka * FEAT + col0 + lmod];
        b.y = W[(size_t)(ka + 1) * FEAT + col0 + lmod];
        acc = __builtin_amdgcn_wmma_f32_16x16x4_f32(
            /*neg_a=*/false, a, /*neg_b=*/false, b,
            /*c_mod=*/(short)0, acc, /*reuse_a=*/false, /*reuse_b=*/false);
    }

    const int mbase = row0 + (hi ? 8 : 0);
    #pragma unroll
    for (int v = 0; v < 8; ++v) {
        int r = mbase + v;
        if (r < n) h[(size_t)r * FEAT + col0 + lmod] = acc[v];
    }
}

// ---------------------------------------------------------------------------
// Kernel 2: per-node attention logits a_src/a_dst [n, H] (one wave per node).
// Lane l covers channel l of every head; full-wave shfl_xor reduction per head.
// ---------------------------------------------------------------------------
__global__ __launch_bounds__(256) void node_logits(const float* __restrict__ h,
                                                   const float* __restrict__ att_src,
                                                   const float* __restrict__ att_dst,
                                                   float* __restrict__ a_src,
                                                   float* __restrict__ a_dst, int n) {
    const int lane   = threadIdx.x & 31;
    const int waveid = (blockIdx.x * blockDim.x + threadIdx.x) >> 5;
    const int nwaves = (gridDim.x * blockDim.x) >> 5;
    for (int node = waveid; node < n; node += nwaves) {
        float ps[HNUM], pd[HNUM];
        #pragma unroll
        for (int hd = 0; hd < HNUM; ++hd) {
            float v = h[(size_t)node * FEAT + hd * CNUM + lane];
            ps[hd] = v * att_src[hd * CNUM + lane];
            pd[hd] = v * att_dst[hd * CNUM + lane];
        }
        #pragma unroll
        for (int off = 16; off > 0; off >>= 1) {
            #pragma unroll
            for (int hd = 0; hd < HNUM; ++hd) {
                ps[hd] += __shfl_xor(ps[hd], off, 32);
                pd[hd] += __shfl_xor(pd[hd], off, 32);
            }
        }
        if (lane == 0) {
            #pragma unroll
            for (int hd = 0; hd < HNUM; ++hd) {
                a_src[node * HNUM + hd] = ps[hd];
                a_dst[node * HNUM + hd] = pd[hd];
            }
        }
    }
}

// ---------------------------------------------------------------------------
// Init kernels (d_ws is poisoned; must re-init every call).
// ---------------------------------------------------------------------------
__global__ void init_out_bias(float* __restrict__ out, const float* __restrict__ bias,
                              int total) {
    int i = blockIdx.x * blockDim.x + threadIdx.x;
    if (i < total) out[i] = bias[i & (FEAT - 1)];
}
__global__ void init_stats(int* __restrict__ mkey, float* __restrict__ ssum,
                           float* __restrict__ sums, float* __restrict__ sumsq,
                           int nh) {
    int i = blockIdx.x * blockDim.x + threadIdx.x;
    if (i < nh) { mkey[i] = INT_MIN; ssum[i] = 0.f; }
    if (i < FEAT) { sums[i] = 0.f; sumsq[i] = 0.f; }
}

// ---------------------------------------------------------------------------
// Kernel 3: segment max of LeakyReLU logits (one thread per edge*head).
// Self-loops are edges with idx >= E.
// ---------------------------------------------------------------------------
__global__ void edge_max(const long long* __restrict__ ei,
                         const float* __restrict__ a_src,
                         const float* __restrict__ a_dst,
                         int* __restrict__ mkey, int E, int n) {
    int gid = blockIdx.x * blockDim.x + threadIdx.x;
    int total = (E + n) * HNUM;
    if (gid >= total) return;
    int idx = gid >> 2, hd = gid & 3;
    int src, dst;
    if (idx < E) { src = (int)ei[idx]; dst = (int)ei[(size_t)E + idx]; }
    else         { src = dst = idx - E; }
    float z = lrelu(a_src[src * HNUM + hd] + a_dst[dst * HNUM + hd]);
    atomicMax(&mkey[dst * HNUM + hd], ordkey(z));
}

// Kernel 4: segment sum of exp(e - m)
__global__ void edge_sum(const long long* __restrict__ ei,
                         const float* __restrict__ a_src,
                         const float* __restrict__ a_dst,
                         const int* __restrict__ mkey,
                         float* __restrict__ ssum, int E, int n) {
    int gid = blockIdx.x * blockDim.x + threadIdx.x;
    int total = (E + n) * HNUM;
    if (gid >= total) return;
    int idx = gid >> 2, hd = gid & 3;
    int src, dst;
    if (idx < E) { src = (int)ei[idx]; dst = (int)ei[(size_t)E + idx]; }
    else         { src = dst = idx - E; }
    float z = lrelu(a_src[src * HNUM + hd] + a_dst[dst * HNUM + hd]);
    float m = ordval(mkey[dst * HNUM + hd]);
    atomicAdd(&ssum[dst * HNUM + hd], __expf(z - m));
}

// ---------------------------------------------------------------------------
// Kernel 5: weighted scatter-add. One wave per edge; lane l owns channels
// [4l, 4l+4) -> head = l>>3 is uniform within the 4-channel group.
// ---------------------------------------------------------------------------
__global__ __launch_bounds__(256) void edge_agg(const long long* __restrict__ ei,
                                                const float* __restrict__ a_src,
                                                const float* __restrict__ a_dst,
                                                const int* __restrict__ mkey,
                                                const float* __restrict__ ssum,
                                                const float* __restrict__ h,
                                                float* __restrict__ out, int E, int n) {
    const int edge = blockIdx.x * 8 + (threadIdx.x >> 5);
    const int lane = threadIdx.x & 31;
    const int total = E + n;
    if (edge >= total) return;
    int src, dst;
    if (edge < E) { src = (int)ei[edge]; dst = (int)ei[(size_t)E + edge]; }
    else          { src = dst = edge - E; }
    const int hd = lane >> 3;
    float z = lrelu(a_src[src * HNUM + hd] + a_dst[dst * HNUM + hd]);
    float m = ordval(mkey[dst * HNUM + hd]);
    float alpha = __expf(z - m) / (ssum[dst * HNUM + hd] + 1e-16f);
    float4 hv = *(const float4*)(h + (size_t)src * FEAT + lane * 4);
    float* o = out + (size_t)dst * FEAT + lane * 4;
    atomicAdd(o + 0, hv.x * alpha);
    atomicAdd(o + 1, hv.y * alpha);
    atomicAdd(o + 2, hv.z * alpha);
    atomicAdd(o + 3, hv.w * alpha);
}

// ---------------------------------------------------------------------------
// Kernel 6: BN column statistics (coalesced row-major sweep, LDS fold, atomics)
// ---------------------------------------------------------------------------
__global__ __launch_bounds__(256) void bn_stats(const float* __restrict__ out,
                                                float* __restrict__ sums,
                                                float* __restrict__ sumsq, int n) {
    const int col   = threadIdx.x & (FEAT - 1);
    const int rhalf = threadIdx.x >> 7;     // 0 or 1: 2 rows per block iter
    float s = 0.f, q = 0.f;
    for (int r = blockIdx.x * 2 + rhalf; r < n; r += gridDim.x * 2) {
        float v = out[(size_t)r * FEAT + col];
        s += v; q += v * v;
    }
    __shared__ float sh[256], shq[256];
    sh[threadIdx.x] = s; shq[threadIdx.x] = q;
    __syncthreads();
    if (rhalf == 0) {
        s += sh[col + 128]; q += shq[col + 128];
        atomicAdd(&sums[col], s);
        atomicAdd(&sumsq[col], q);
    }
}

// Kernel 7: in-place BN + residual + ReLU on d_out
__global__ void finalize(float* __restrict__ out, const float* __restrict__ x,
                         const float* __restrict__ sums, const float* __restrict__ sumsq,
                         const float* __restrict__ gamma, const float* __restrict__ beta,
                         int n) {
    int i = blockIdx.x * blockDim.x + threadIdx.x;
    int total = n * FEAT;
    if (i >= total) return;
    int c = i & (FEAT - 1);
    float inv_n = 1.f / (float)n;
    float mean = sums[c] * inv_n;
    float var  = sumsq[c] * inv_n - mean * mean;   // biased var (jnp.var default)
    float v = gamma[c] * (out[i] - mean) * rsqrtf(var + BN_EPS) + beta[c] + x[i];
    out[i] = v > 0.f ? v : 0.f;
}

// ---------------------------------------------------------------------------
extern "C" void kernel_launch(void* const* d_in, const int* in_sizes, int n_in,
                              void* d_out, int out_size, void* d_ws, size_t ws_size,
                              hipStream_t stream) {
    const float*     x        = (const float*)d_in[0];
    const long long* ei       = (const long long*)d_in[1];   // int64 [2, E]
    const float*     W        = (const float*)d_in[2];
    const float*     att_src  = (const float*)d_in[3];
    const float*     att_dst  = (const float*)d_in[4];
    const float*     gat_bias = (const float*)d_in[5];
    const float*     bn_gamma = (const float*)d_in[6];
    const float*     bn_beta  = (const float*)d_in[7];
    float*           out      = (float*)d_out;

    const int n = in_sizes[0] / FEAT;
    const int E = in_sizes[1] / 2;
    const int nh = n * HNUM;

    // Workspace carve-up (floats/ints, 4B aligned)
    char* ws = (char*)d_ws;
    float* h      = (float*)ws;                  ws += (size_t)n * FEAT * sizeof(float);
    float* a_src  = (float*)ws;                  ws += (size_t)nh * sizeof(float);
    float* a_dst  = (float*)ws;                  ws += (size_t)nh * sizeof(float);
    int*   mkey   = (int*)ws;                    ws += (size_t)nh * sizeof(int);
    float* ssum   = (float*)ws;                  ws += (size_t)nh * sizeof(float);
    float* sums   = (float*)ws;                  ws += FEAT * sizeof(float);
    float* sumsq  = (float*)ws;                  /* end */

    // 1) GEMM h = x @ W (WMMA f32 16x16x4)
    gemm_wmma_f32<<<(n + 15) / 16, 256, 0, stream>>>(x, W, h, n);

    // 2) per-node logits
    node_logits<<<1024, 256, 0, stream>>>(h, att_src, att_dst, a_src, a_dst, n);

    // 3) init accumulators (ws + d_out are poisoned between calls)
    init_out_bias<<<((size_t)n * FEAT + 255) / 256, 256, 0, stream>>>(out, gat_bias, n * FEAT);
    init_stats<<<(nh + 255) / 256, 256, 0, stream>>>(mkey, ssum, sums, sumsq, nh);

    // 4) segment softmax passes
    const int totEH = (E + n) * HNUM;
    edge_max<<<(totEH + 255) / 256, 256, 0, stream>>>(ei, a_src, a_dst, mkey, E, n);
    edge_sum<<<(totEH + 255) / 256, 256, 0, stream>>>(ei, a_src, a_dst, mkey, ssum, E, n);

    // 5) weighted scatter aggregation (1 wave / edge)
    edge_agg<<<((E + n) + 7) / 8, 256, 0, stream>>>(ei, a_src, a_dst, mkey, ssum, h, out, E, n);

    // 6) BN stats + 7) finalize in place
    bn_stats<<<1024, 256, 0, stream>>>(out, sums, sumsq, n);
    finalize<<<((size_t)n * FEAT + 255) / 256, 256, 0, stream>>>(out, x, sums, sumsq,
                                                                 bn_gamma, bn_beta, n);
}